// GIN_52407190946419
// MI455X (gfx1250) — compile-verified
//
#include <hip/hip_runtime.h>

#define D 128

typedef __attribute__((ext_vector_type(2))) float v2f;
typedef __attribute__((ext_vector_type(8))) float v8f;

// ---------------------------------------------------------------------------
// Kernel 1: h = x   (so the edge scatter produces h = x + sum_neighbors)
// ---------------------------------------------------------------------------
__global__ void gin_init_copy(const float* __restrict__ x,
                              float* __restrict__ h, long n4) {
    long i = (long)blockIdx.x * blockDim.x + threadIdx.x;
    if (i < n4) ((float4*)h)[i] = ((const float4*)x)[i];
}

// ---------------------------------------------------------------------------
// Kernel 2: edge scatter. One 32-lane wave per edge; each lane moves one
// float4 (32 lanes * 16B = full 512B feature row, coalesced gather) and does
// 4 native f32 atomic adds. x (51MB) and h (51MB) both fit in the 192MB L2,
// so gathers and atomics are serviced at L2 rates, not HBM.
// ---------------------------------------------------------------------------
__global__ void gin_edge_scatter(const float* __restrict__ x,
                                 const int* __restrict__ ei,
                                 float* __restrict__ h, int E) {
    long t = (long)blockIdx.x * blockDim.x + threadIdx.x;
    int e    = (int)(t >> 5);
    int lane = (int)(t & 31);
    if (e >= E) return;
    int src = ei[e];
    int dst = ei[(long)E + e];
    float4 v = ((const float4*)(x + (long)src * D))[lane];
    float* hp = h + (long)dst * D + lane * 4;
    unsafeAtomicAdd(hp + 0, v.x);   // -> global_atomic_add_f32 (no CAS loop)
    unsafeAtomicAdd(hp + 1, v.y);
    unsafeAtomicAdd(hp + 2, v.z);
    unsafeAtomicAdd(hp + 3, v.w);
}

// ---------------------------------------------------------------------------
// Kernel 3: Hout = relu(Hin @ W + b), M x 128 @ 128 x 128, f32 WMMA.
// Block = 256 threads = 8 waves; each wave computes one 16-row strip across
// all 128 output columns (8 WMMA 16x16 tiles), K-loop in steps of 4 using
// V_WMMA_F32_16X16X4_F32. W (64KB) is staged in LDS once per block.
//
// The wave's whole A strip (256B/lane = 32 x float2 = 64 VGPRs) is preloaded
// into registers before the K loop: 32 global_load_b64 in flight, consumed
// in order, so the K loop never does a cold s_wait_loadcnt 0.
//
// Fragment layouts per CDNA5 ISA 7.12.2 (wave32):
//   A 16x4:  lane l -> row l%16, holds K = {kb, kb+1}, kb = 2*(l>>4)
//   B 4x16:  lane l -> col l%16, b.x = W[kk+kb][n], b.y = W[kk+kb+1][n]
//   C 16x16: vgpr v, lane l -> row v + 8*(l>>4), col l%16
// ---------------------------------------------------------------------------
__global__ __launch_bounds__(256) void gin_gemm_bias_relu(
        const float* __restrict__ Hin, const float* __restrict__ W,
        const float* __restrict__ bias, float* __restrict__ Hout, int M) {
    __shared__ float Wl[D * D];   // 64 KB

    // Cooperative load of W into LDS (coalesced float4).
    for (int i = threadIdx.x; i < (D * D) / 4; i += 256)
        ((float4*)Wl)[i] = ((const float4*)W)[i];
    __syncthreads();

    const int lane = threadIdx.x & 31;
    const int wv   = threadIdx.x >> 5;
    const int row0 = blockIdx.x * 128 + wv * 16;
    if (row0 >= M) return;             // wave-uniform: EXEC stays all-ones

    const int m    = lane & 15;        // A row / C col within tile
    const int half = lane >> 4;
    const int kb   = half * 2;         // K sub-offset for this lane half

    const float* hrow = Hin + (long)(row0 + m) * D + kb;

    // Preload the entire A strip: 32 independent 8B loads in flight.
    v2f afrag[32];
#pragma unroll
    for (int t = 0; t < 32; ++t)
        afrag[t] = *(const v2f*)(hrow + t * 4);

    v8f acc[8] = {};

#pragma unroll
    for (int kt = 0; kt < 32; ++kt) {
        const int kk = kt * 4;
        v2f a = afrag[kt];
        const float* wr0 = &Wl[(kk + kb) * D + m];
        const float* wr1 = wr0 + D;
#pragma unroll
        for (int j = 0; j < 8; ++j) {
            v2f b;
            b.x = wr0[j * 16];
            b.y = wr1[j * 16];
            acc[j] = __builtin_amdgcn_wmma_f32_16x16x4_f32(
                false, a, false, b, (short)0, acc[j], false, false);
        }
    }

    // Epilogue: bias + ReLU, store via C/D layout (rows half-coalesced).
#pragma unroll
    for (int j = 0; j < 8; ++j) {
        int   col = j * 16 + m;
        float bj  = bias[col];
#pragma unroll
        for (int v = 0; v < 8; ++v) {
            int   r   = row0 + v + 8 * half;
            float val = acc[j][v] + bj;
            Hout[(long)r * D + col] = val > 0.f ? val : 0.f;
        }
    }
}

// ---------------------------------------------------------------------------
// Launch: ws holds ONE [M x 128] f32 buffer (51.2 MB); t1 lives in d_out.
//   h  = x + agg          -> ws
//   t1 = relu(h  @W1+b1)  -> d_out
//   t2 = relu(t1 @W2+b2)  -> ws      (h no longer needed)
//   y  = relu(t2 @W3+b3)  -> d_out
// ---------------------------------------------------------------------------
extern "C" void kernel_launch(void* const* d_in, const int* in_sizes, int n_in,
                              void* d_out, int out_size, void* d_ws, size_t ws_size,
                              hipStream_t stream) {
    const float* x  = (const float*)d_in[0];
    const float* W1 = (const float*)d_in[1];
    const float* b1 = (const float*)d_in[2];
    const float* W2 = (const float*)d_in[3];
    const float* b2 = (const float*)d_in[4];
    const float* W3 = (const float*)d_in[5];
    const float* b3 = (const float*)d_in[6];
    const int*   ei = (const int*)d_in[7];

    const int E = in_sizes[7] / 2;
    const int M = out_size / D;

    float* hbuf = (float*)d_ws;
    float* out  = (float*)d_out;

    long n4 = (long)M * D / 4;
    gin_init_copy<<<(int)((n4 + 255) / 256), 256, 0, stream>>>(x, hbuf, n4);

    long tE = (long)E * 32;
    gin_edge_scatter<<<(int)((tE + 255) / 256), 256, 0, stream>>>(x, ei, hbuf, E);

    int gBlocks = (M + 127) / 128;
    gin_gemm_bias_relu<<<gBlocks, 256, 0, stream>>>(hbuf, W1, b1, out,  M);
    gin_gemm_bias_relu<<<gBlocks, 256, 0, stream>>>(out,  W2, b2, hbuf, M);
    gin_gemm_bias_relu<<<gBlocks, 256, 0, stream>>>(hbuf, W3, b3, out,  M);
}